// VectorQuantizedBottleneck_13941463842923
// MI455X (gfx1250) — compile-verified
//
#include <hip/hip_runtime.h>
#include <stdint.h>

#define N_ROWS 2048
#define N_DIM  256
#define N_CODE 512

#define BD  16   // dims per block
#define BN  128  // rows per block
#define TY  16   // row-groups (threads in y)
#define RPT 8    // rows per thread = BN/TY

typedef uint32_t v4u __attribute__((ext_vector_type(4)));
typedef int      v8i __attribute__((ext_vector_type(8)));
typedef int      v4i __attribute__((ext_vector_type(4)));

__global__ __launch_bounds__(256) void vq_main_kernel(
    const float* __restrict__ enc, const float* __restrict__ emb,
    float* __restrict__ out, float* __restrict__ partial) {
  __shared__ __align__(16) float lds_emb[BD * N_CODE];  // 32 KB
  __shared__ float lds_red[8];

  const int tid = threadIdx.x;
  const int tx  = tid & (BD - 1);   // dim within tile
  const int ty  = tid >> 4;         // row group
  const int d0  = blockIdx.x * BD;
  const int n0  = blockIdx.y * BN;

  // ---- Stage emb[d0:d0+16, 0:512] (32 KB, contiguous) into LDS via TDM ----
#if defined(__gfx1250__) && __has_builtin(__builtin_amdgcn_tensor_load_to_lds)
  if (tid < 32) {  // wave 0 issues one TDM op (EXEC==0 waves branch over)
    uint64_t gaddr = (uint64_t)(uintptr_t)(emb + (size_t)d0 * N_CODE);
    uint32_t laddr = (uint32_t)(uintptr_t)(void*)lds_emb;  // LDS offset = low 32 bits
    v4u g0;
    g0[0] = 1u;                                   // count=1, user descriptor
    g0[1] = laddr;                                // lds_addr (bytes)
    g0[2] = (uint32_t)gaddr;                      // global_addr[31:0] (tile start)
    g0[3] = ((uint32_t)(gaddr >> 32) & 0x01FFFFFFu) | (2u << 30);  // addr[56:32], type=2
    v8i g1;
    g1[0] = (int)(2u << 16);                      // data_size=2 (4 bytes), mask=0, no pad
    g1[1] = (int)((uint32_t)N_CODE << 16);        // tensor_dim0[15:0]=512
    g1[2] = (int)((uint32_t)BD << 16);            // dim0 hi=0 | tensor_dim1[15:0]=16
    g1[3] = (int)((uint32_t)N_CODE << 16);        // dim1 hi=0 | tile_dim0=512
    g1[4] = BD;                                   // tile_dim1=16 | tile_dim2=0
    g1[5] = N_CODE;                               // tensor_dim0_stride lo32 = 512
    g1[6] = 0;                                    // stride0 hi | tensor_dim1_stride lo
    g1[7] = 0;
    v4i gz4 = {0, 0, 0, 0};
#if __clang_major__ >= 23
    v8i gz8 = {0, 0, 0, 0, 0, 0, 0, 0};
    __builtin_amdgcn_tensor_load_to_lds(g0, g1, gz4, gz4, gz8, 0);  // 6-arg form
#else
    __builtin_amdgcn_tensor_load_to_lds(g0, g1, gz4, gz4, 0);       // 5-arg form
#endif
    __builtin_amdgcn_s_wait_tensorcnt(0);
  }
#else
  for (int i = tid; i < BD * N_CODE; i += 256)
    lds_emb[i] = emb[(size_t)d0 * N_CODE + i];
#endif

  // ---- Load this thread's 8 encoded scalars (overlaps with TDM) ----
  float x[RPT], bA[RPT], bV[RPT];
#pragma unroll
  for (int r = 0; r < RPT; ++r) {
    int n = n0 + ty + TY * r;
    x[r]  = enc[(size_t)n * N_DIM + d0 + tx];
    bA[r] = 3.0e38f;
    bV[r] = 0.0f;
  }
  __syncthreads();

  // ---- Argmin scan: rotated start per tx -> conflict-free LDS banks ----
  const float* row = &lds_emb[tx * N_CODE];
  int kk = (4 * tx) & (N_CODE - 1);
  for (int it = 0; it < N_CODE / 4; ++it) {
    const float4 e = *(const float4*)(row + kk);
    kk = (kk + 4) & (N_CODE - 1);
#pragma unroll
    for (int r = 0; r < RPT; ++r) {
      float a0 = fabsf(e.x - x[r]); bool c0 = a0 < bA[r];
      bA[r] = c0 ? a0 : bA[r];      bV[r] = c0 ? e.x : bV[r];
      float a1 = fabsf(e.y - x[r]); bool c1 = a1 < bA[r];
      bA[r] = c1 ? a1 : bA[r];      bV[r] = c1 ? e.y : bV[r];
      float a2 = fabsf(e.z - x[r]); bool c2 = a2 < bA[r];
      bA[r] = c2 ? a2 : bA[r];      bV[r] = c2 ? e.z : bV[r];
      float a3 = fabsf(e.w - x[r]); bool c3 = a3 < bA[r];
      bA[r] = c3 ? a3 : bA[r];      bV[r] = c3 ? e.w : bV[r];
    }
  }

  // ---- Write latent values, accumulate local loss ----
  float ls = 0.0f;
#pragma unroll
  for (int r = 0; r < RPT; ++r) {
    int n = n0 + ty + TY * r;
    out[(size_t)n * N_DIM + d0 + tx] = bV[r];
    ls += bA[r] * bA[r];
  }

  // ---- Deterministic block reduction ----
  for (int off = 16; off > 0; off >>= 1) ls += __shfl_xor(ls, off, 32);
  if ((tid & 31) == 0) lds_red[tid >> 5] = ls;
  __syncthreads();
  if (tid == 0) {
    float s = 0.0f;
#pragma unroll
    for (int w = 0; w < 8; ++w) s += lds_red[w];
    partial[blockIdx.y * gridDim.x + blockIdx.x] = s;
  }
}

__global__ __launch_bounds__(256) void vq_reduce_kernel(
    const float* __restrict__ partial, float* __restrict__ loss, int n_part) {
  __shared__ float red[8];
  float s = 0.0f;
  for (int i = threadIdx.x; i < n_part; i += 256) s += partial[i];
  for (int off = 16; off > 0; off >>= 1) s += __shfl_xor(s, off, 32);
  if ((threadIdx.x & 31) == 0) red[threadIdx.x >> 5] = s;
  __syncthreads();
  if (threadIdx.x == 0) {
    float t = 0.0f;
#pragma unroll
    for (int w = 0; w < 8; ++w) t += red[w];
    *loss = 2.0f * t;  // vq_loss + commitment_loss (numerically equal)
  }
}

extern "C" void kernel_launch(void* const* d_in, const int* in_sizes, int n_in,
                              void* d_out, int out_size, void* d_ws, size_t ws_size,
                              hipStream_t stream) {
  const float* enc = (const float*)d_in[0];  // (2048, 256)
  const float* emb = (const float*)d_in[1];  // (1, 256, 512) -> (256, 512)
  float* out     = (float*)d_out;            // 2048*256 latent + 1 loss
  float* partial = (float*)d_ws;             // 256 partial sums

  dim3 grid(N_DIM / BD, N_ROWS / BN);        // 16 x 16 = 256 blocks
  vq_main_kernel<<<grid, 256, 0, stream>>>(enc, emb, out, partial);
  vq_reduce_kernel<<<1, 256, 0, stream>>>(partial, out + (size_t)N_ROWS * N_DIM,
                                          (N_DIM / BD) * (N_ROWS / BN));
}